// WectLayer_65403761983812
// MI455X (gfx1250) — compile-verified
//
#include <hip/hip_runtime.h>
#include <hip/hip_bf16.h>

typedef __attribute__((ext_vector_type(2))) float v2f;
typedef __attribute__((ext_vector_type(8))) float v8f;

#define SHARP 500.0f

// ---------------------------------------------------------------------------
// Kernel 1: nh[N,32] = x[N,3] @ v[3,32] via V_WMMA_F32_16X16X4_F32 (K padded to 4).
// One wave per 16-row tile; two WMMAs per wave (col blocks 0-15 and 16-31).
//
// f32 A 16x4 layout (ISA 7.12.2): lanes 0-15 -> M=lane, VGPR0=K0, VGPR1=K1;
//                                 lanes 16-31 -> M=lane-16, VGPR0=K2, VGPR1=K3.
// B 4x16 mirrored (K striped across lane halves). C/D: VGPR j holds
// row j (lanes 0-15) / row 8+j (lanes 16-31), col = lane%16.
//
// Bounds handling is tile-uniform: full tiles take a branch-free path
// (no per-row EXEC juggling); only a ragged last tile is guarded.
// ---------------------------------------------------------------------------
__global__ void heights_wmma_kernel(const float* __restrict__ x,
                                    const float* __restrict__ v,
                                    float* __restrict__ nh,
                                    int N, int tiles) {
    const int lane = threadIdx.x & 31;
    const int wave = (int)((blockIdx.x * blockDim.x + threadIdx.x) >> 5);
    if (wave >= tiles) return;                 // wave-uniform: EXEC stays full
    const int row0 = wave * 16;
    const int half = lane & 15;
    const bool hi  = lane >= 16;
    const bool fullTile = (row0 + 16 <= N);    // wave-uniform

    // A matrix: x rows, K in {0,1,2, pad 0}. Branch-free selects in fast path.
    const int arow = row0 + half;
    v2f a;
    if (fullTile) {
        // unconditional loads (all in bounds), lane-select instead of exec mask
        const float xk02 = x[arow * 3 + (hi ? 2 : 0)];
        const float xk1  = x[arow * 3 + 1];
        a.x = xk02;
        a.y = hi ? 0.0f : xk1;
    } else {
        a.x = (arow < N) ? x[arow * 3 + (hi ? 2 : 0)] : 0.0f;
        a.y = (!hi && arow < N) ? x[arow * 3 + 1] : 0.0f;
    }

#pragma unroll
    for (int cb = 0; cb < 2; ++cb) {
        const int col = cb * 16 + half;
        v2f b;
        b.x = v[(hi ? 2 : 0) * 32 + col];      // K0 | K2
        b.y = hi ? 0.0f : v[32 + col];         // K1 | K3=0

        v8f c = {};
        c = __builtin_amdgcn_wmma_f32_16x16x4_f32(
                /*neg_a=*/false, a, /*neg_b=*/false, b,
                /*c_mod=*/(short)0, c, /*reuse_a=*/false, /*reuse_b=*/false);

        const int rbase = row0 + (hi ? 8 : 0);
        if (fullTile) {
#pragma unroll
            for (int j = 0; j < 8; ++j)
                nh[(rbase + j) * 32 + col] = c[j];
        } else {
#pragma unroll
            for (int j = 0; j < 8; ++j) {
                const int r = rbase + j;
                if (r < N) nh[r * 32 + col] = c[j];
            }
        }
    }
}

// ---------------------------------------------------------------------------
// Kernel 2: fused weighted-ECC accumulation for nodes(+), edges(-), faces(+).
// Each block keeps the FULL [B,32,32] accumulator (256 KB) in LDS (320 KB/WGP),
// does ds_add_f32 atomics (lane==theta -> distinct banks within a wave), and
// flushes once with global fp32 atomics. Combined grid-stride over N+E+F.
// ---------------------------------------------------------------------------
__global__ void wecc_accum_kernel(const float* __restrict__ nh,
                                  const float* __restrict__ nw,
                                  const float* __restrict__ lin_g,
                                  const int*   __restrict__ edge_index,
                                  const int*   __restrict__ face,
                                  const int*   __restrict__ batch,
                                  float* __restrict__ out,
                                  int N, int E, int F, int B) {
    extern __shared__ float lds[];             // B*32*32 floats
    const int BST  = B << 10;                  // B*1024
    const int lane = threadIdx.x & 31;         // lane == theta index t

    // zero the block-private accumulator
    for (int i = threadIdx.x; i < BST; i += blockDim.x) lds[i] = 0.0f;
    __syncthreads();

    // pre-scaled thresholds in registers: rlin[s] = 500*lin[s]
    float rlin[32];
#pragma unroll
    for (int s = 0; s < 32; ++s) rlin[s] = SHARP * lin_g[s];

    const int wave   = (int)((blockIdx.x * blockDim.x + threadIdx.x) >> 5);
    const int nWaves = (int)((gridDim.x * blockDim.x) >> 5);
    const int Mtot   = N + E + F;

    for (int m = wave; m < Mtot; m += nWaves) {
        float h, wsgn;
        int b;
        if (m < N) {                                   // node simplex, sign +
            h    = nh[m * 32 + lane];
            wsgn = nw[m];
            b    = batch[m];
        } else if (m < N + E) {                        // edge simplex, sign -
            const int e  = m - N;
            const int i0 = edge_index[e];
            const int i1 = edge_index[E + e];
            h    = fminf(nh[i0 * 32 + lane], nh[i1 * 32 + lane]);
            wsgn = -fmaxf(nw[i0], nw[i1]);
            b    = batch[i0];
        } else {                                       // face simplex, sign +
            const int f  = m - N - E;
            const int f0 = face[f];
            const int f1 = face[F + f];
            const int f2 = face[2 * F + f];
            h    = fminf(fminf(nh[f0 * 32 + lane], nh[f1 * 32 + lane]),
                         nh[f2 * 32 + lane]);
            wsgn = fmaxf(fmaxf(nw[f0], nw[f1]), nw[f2]);
            b    = batch[f0];
        }

        const float h500 = SHARP * h;
        float* __restrict__ row = lds + (b << 10) + lane;
#pragma unroll
        for (int s = 0; s < 32; ++s) {
            // sigmoid(500*(lin[s]-h)) = 1 / (1 + exp(500*h - 500*lin[s]))
            const float e   = __expf(h500 - rlin[s]);
            const float sig = __builtin_amdgcn_rcpf(1.0f + e);
            unsafeAtomicAdd(row + (s << 5), wsgn * sig);   // ds_add_f32
        }
    }

    __syncthreads();
    // single flush of the block-private accumulator
    for (int i = threadIdx.x; i < BST; i += blockDim.x) {
        const float vaccum = lds[i];
        if (vaccum != 0.0f) unsafeAtomicAdd(&out[i], vaccum);  // global_atomic_add_f32
    }
}

extern "C" void kernel_launch(void* const* d_in, const int* in_sizes, int n_in,
                              void* d_out, int out_size, void* d_ws, size_t ws_size,
                              hipStream_t stream) {
    const float* x          = (const float*)d_in[0];   // [N,3]
    const float* nw         = (const float*)d_in[1];   // [N]
    const float* v          = (const float*)d_in[2];   // [3,32]
    const float* lin        = (const float*)d_in[3];   // [32]
    const int*   edge_index = (const int*)d_in[4];     // [2,E]
    const int*   face       = (const int*)d_in[5];     // [3,F]
    const int*   batch      = (const int*)d_in[6];     // [N]
    float*       out        = (float*)d_out;           // [B,32,32]

    const int N = in_sizes[1];
    const int E = in_sizes[4] / 2;
    const int F = in_sizes[5] / 3;
    const int B = out_size / (32 * 32);

    float* nh = (float*)d_ws;                          // N*32 floats scratch

    // d_out is poisoned by the harness; we accumulate atomically, so zero it.
    hipMemsetAsync(d_out, 0, (size_t)out_size * sizeof(float), stream);

    // Kernel 1: WMMA heights. One wave per 16-row tile, 8 waves per block.
    const int tiles   = (N + 15) / 16;
    const int blocks1 = (tiles + 7) / 8;
    heights_wmma_kernel<<<blocks1, 256, 0, stream>>>(x, v, nh, N, tiles);

    // Kernel 2: fused ECC accumulation. 1024 threads (32 waves) per block,
    // full [B,S,T] accumulator in LDS (256 KB of the WGP's 320 KB).
    const int    blocks2  = 256;
    const size_t ldsBytes = (size_t)B * 1024 * sizeof(float);  // 256 KB for B=64
    wecc_accum_kernel<<<blocks2, 1024, ldsBytes, stream>>>(
        nh, nw, lin, edge_index, face, batch, out, N, E, F, B);
}